// Tokenizer_5892695130625
// MI455X (gfx1250) — compile-verified
//
#include <hip/hip_runtime.h>

typedef float v2f __attribute__((ext_vector_type(2)));
typedef float v8f __attribute__((ext_vector_type(8)));

#define B_   8
#define S_   19
#define SP   32          // S padded to 2x16 M-tiles
#define F_   512
#define HC_  64
#define HW_  4096        // 64*64
#define HS_  512
#define KC_  4           // K-split chunks in seg-sum GEMM
#define KLEN_ (HW_ / KC_)  // 1024

// ---------------------------------------------------------------------------
// Kernel 1: build padded mask matrix (f32 0/1) + per-(b,s) counts.
// One 256-thread block per (b, s_padded). Rows s>=19 are zeroed.
// ---------------------------------------------------------------------------
__global__ void build_mask_kernel(const int* __restrict__ segmap,
                                  float* __restrict__ maskf,
                                  float* __restrict__ counts) {
    const int bs  = blockIdx.x;            // b*32 + s
    const int b   = bs >> 5;
    const int s   = bs & 31;
    const int tid = threadIdx.x;

    __shared__ int red[256];
    int cnt = 0;
    float* mrow = maskf + (size_t)bs * HW_;

    if (s < S_) {
        const int* seg = segmap + (size_t)(b * S_ + s) * (HS_ * HS_);
        for (int i = tid; i < HW_; i += 256) {
            const int h = i >> 6;
            const int w = i & 63;
            // nearest: (i*512)//64 == i*8
            const int v = seg[(h * 8) * HS_ + (w * 8)];
            mrow[i] = (v != 0) ? 1.0f : 0.0f;
            cnt += (v != 0) ? 1 : 0;
        }
    } else {
        for (int i = tid; i < HW_; i += 256) mrow[i] = 0.0f;
    }

    red[tid] = cnt;
    __syncthreads();
    for (int off = 128; off > 0; off >>= 1) {
        if (tid < off) red[tid] += red[tid + off];
        __syncthreads();
    }
    if (tid == 0) counts[bs] = (float)red[0];
}

// ---------------------------------------------------------------------------
// Kernel 2: segment-sum GEMM via V_WMMA_F32_16X16X4_F32.
// D[M=s, N=f] = sum_k mask[s,k] * codes[f,k], per batch.
// 1024 waves: 8 b * 2 m-tiles * 16 f-tile-pairs * 4 K-chunks.
// Each wave: A = mask rows (b64/lane), B = codes cols (b64/lane), 2 C tiles.
// ---------------------------------------------------------------------------
__global__ void seg_gemm_kernel(const float* __restrict__ codes,
                                const float* __restrict__ maskf,
                                float* __restrict__ partial) {
    const int wg   = blockIdx.x * 8 + (threadIdx.x >> 5);  // global wave id
    const int kc   = wg & 3;
    const int ntg  = (wg >> 2) & 15;    // pair of f-tiles
    const int mt   = (wg >> 6) & 1;
    const int b    = wg >> 7;

    const int lane = threadIdx.x & 31;
    const int half = lane >> 4;         // 0: K+{0,1}, 1: K+{2,3}
    const int lm   = lane & 15;

    const int srow = mt * 16 + lm;          // A row for this lane
    const int f0   = ntg * 32 + lm;         // B col, tile 0
    const int f1   = f0 + 16;               // B col, tile 1

    const float* Ab  = maskf + (size_t)(b * SP + srow) * HW_;
    const float* Bb0 = codes + (size_t)(b * F_ + f0) * HW_;
    const float* Bb1 = codes + (size_t)(b * F_ + f1) * HW_;

    const int kbeg = kc * KLEN_ + 2 * half;

    v8f c0 = {};
    v8f c1 = {};

#pragma unroll 4
    for (int k = 0; k < KLEN_; k += 4) {
        const v2f a  = *(const v2f*)(Ab  + kbeg + k);
        const v2f b0 = *(const v2f*)(Bb0 + kbeg + k);
        const v2f b1 = *(const v2f*)(Bb1 + kbeg + k);
        c0 = __builtin_amdgcn_wmma_f32_16x16x4_f32(
                 false, a, false, b0, (short)0, c0, false, false);
        c1 = __builtin_amdgcn_wmma_f32_16x16x4_f32(
                 false, a, false, b1, (short)0, c1, false, false);
    }

    // C/D layout: vgpr r, lanes 0-15 -> M=r, lanes 16-31 -> M=r+8; N = lane%16.
    float* pt = partial + (size_t)kc * (B_ * SP * F_);
#pragma unroll
    for (int r = 0; r < 8; ++r) {
        const int sr = mt * 16 + half * 8 + r;
        const size_t o = (size_t)(b * SP + sr) * F_;
        pt[o + f0] = c0[r];
        pt[o + f1] = c1[r];
    }
}

// ---------------------------------------------------------------------------
// Kernel 3: reduce K-chunk partials, divide by count -> mean[b][32][512].
// ---------------------------------------------------------------------------
__global__ void reduce_mean_kernel(const float* __restrict__ partial,
                                   const float* __restrict__ counts,
                                   float* __restrict__ mean) {
    const int i = blockIdx.x * 256 + threadIdx.x;   // over B_*SP*F_ = 131072
    const int n = B_ * SP * F_;
    const int bs = i / F_;
    const float cnt = counts[bs];
    const float v = partial[i] + partial[i + n] + partial[i + 2 * n] + partial[i + 3 * n];
    mean[i] = (cnt > 0.0f) ? (v / cnt) : 0.0f;
}

// ---------------------------------------------------------------------------
// Kernel 4: FC GEMM: out[b,s,fo] = sum_f mean[b,s,f] * fc_w[fo,f] + fc_b[fo].
// 512 waves: 8 b * 2 m-tiles * 32 n-tiles. K = 512.
// ---------------------------------------------------------------------------
__global__ void fc_gemm_kernel(const float* __restrict__ mean,
                               const float* __restrict__ fc_w,
                               const float* __restrict__ fc_b,
                               float* __restrict__ out) {
    const int wg = blockIdx.x * 8 + (threadIdx.x >> 5);
    const int nt = wg & 31;
    const int mt = (wg >> 5) & 1;
    const int b  = wg >> 6;

    const int lane = threadIdx.x & 31;
    const int half = lane >> 4;
    const int lm   = lane & 15;

    const int srow = mt * 16 + lm;
    const int fo   = nt * 16 + lm;

    const float* Ab = mean + (size_t)(b * SP + srow) * F_;
    const float* Bb = fc_w + (size_t)fo * F_;
    const int koff = 2 * half;

    v8f c = {};

#pragma unroll 4
    for (int k = 0; k < F_; k += 4) {
        const v2f a  = *(const v2f*)(Ab + koff + k);
        const v2f bm = *(const v2f*)(Bb + koff + k);
        c = __builtin_amdgcn_wmma_f32_16x16x4_f32(
                false, a, false, bm, (short)0, c, false, false);
    }

    const float bias = fc_b[fo];
#pragma unroll
    for (int r = 0; r < 8; ++r) {
        const int sr = mt * 16 + half * 8 + r;
        if (sr < S_) {
            out[(size_t)(b * S_ + sr) * F_ + fo] = c[r] + bias;
        }
    }
}

// ---------------------------------------------------------------------------
// Launch. Workspace layout (floats):
//   maskf   : B_*SP*HW_          = 1,048,576
//   counts  : B_*SP              = 256
//   partial : KC_*B_*SP*F_       = 524,288
//   mean    : B_*SP*F_           = 131,072
//   total  ~= 6.8 MB
// ---------------------------------------------------------------------------
extern "C" void kernel_launch(void* const* d_in, const int* in_sizes, int n_in,
                              void* d_out, int out_size, void* d_ws, size_t ws_size,
                              hipStream_t stream) {
    const float* codes  = (const float*)d_in[0];
    const int*   segmap = (const int*)d_in[1];
    const float* fc_w   = (const float*)d_in[2];
    const float* fc_b   = (const float*)d_in[3];
    float* out = (float*)d_out;

    float* maskf   = (float*)d_ws;
    float* counts  = maskf + (size_t)B_ * SP * HW_;
    float* partial = counts + (size_t)B_ * SP;
    float* mean    = partial + (size_t)KC_ * B_ * SP * F_;

    // 1) mask + counts: one block per (b, padded s)
    build_mask_kernel<<<B_ * SP, 256, 0, stream>>>(segmap, maskf, counts);

    // 2) segment-sum GEMM: 1024 waves = 128 blocks * 8 waves
    seg_gemm_kernel<<<128, 256, 0, stream>>>(codes, maskf, partial);

    // 3) reduce partials + mean
    reduce_mean_kernel<<<(B_ * SP * F_) / 256, 256, 0, stream>>>(partial, counts, mean);

    // 4) FC GEMM: 512 waves = 64 blocks * 8 waves
    fc_gemm_kernel<<<64, 256, 0, stream>>>(mean, fc_w, fc_b, out);
}